// MultiReflection_20512763806447
// MI455X (gfx1250) — compile-verified
//
#include <hip/hip_runtime.h>

typedef __attribute__((ext_vector_type(2))) float v2f;
typedef __attribute__((ext_vector_type(8))) float v8f;

#define NBATCH   8192
#define NLAYER   60
#define NBAND    16
#define NCH      48
#define NROWS    62            // NLAYER + 2
#define BPB      2             // batches per block
#define NTHREADS 96            // BPB * NCH = 3 wave32
#define MROWS    (BPB * NROWS) // 124 rows of x_sources per block

__global__ __launch_bounds__(NTHREADS, 1)
void radiative_scan_kernel(const float* __restrict__ x_sources,   // (B, 62, 16)
                           const float* __restrict__ t_full,      // (B, 60, 48)
                           const float* __restrict__ e_split,     // (B, 60, 48, 3)
                           const float* __restrict__ tau,         // (B, 60, 48)
                           const float* __restrict__ x_emis,      // (B,)
                           const float* __restrict__ b2c,         // (16, 48)
                           float* __restrict__ flux_down,         // (B, 61, 48)
                           float* __restrict__ flux_up)           // (B, 61, 48)
{
    __shared__ float sW[NBAND][NCH];    // softmax(b2c) : 3 KB
    __shared__ float sXS[MROWS][NCH];   // xs tile      : ~24 KB

    const int tid = threadIdx.x;

    // ---- Phase 0: W = softmax(bands_to_channels, axis=1) ----
    if (tid < NBAND) {
        const float* wrow = b2c + tid * NCH;
        float mx = -1e30f;
        #pragma unroll
        for (int c = 0; c < NCH; ++c) mx = fmaxf(mx, wrow[c]);
        float s = 0.f;
        #pragma unroll
        for (int c = 0; c < NCH; ++c) { float e = __expf(wrow[c] - mx); sW[tid][c] = e; s += e; }
        float inv = 1.f / s;
        #pragma unroll
        for (int c = 0; c < NCH; ++c) sW[tid][c] *= inv;
    }
    __syncthreads();

    // ---- Phase 1: xs(124x48) = Xsrc(124x16) @ W(16x48) via v_wmma_f32_16x16x4_f32 ----
    {
        const int lane = tid & 31;
        const int nt   = tid >> 5;          // wave id == N tile (0..2)
        const int half = lane >> 4;         // 0 | 1
        const int lm   = lane & 15;
        const int ncol = nt * 16 + lm;

        // B fragments for the 4 K-steps: B is 4x16 f32, rows striped across lane halves
        v2f bfrag[4];
        #pragma unroll
        for (int kk = 0; kk < 4; ++kk) {
            const int krow = 4 * kk + 2 * half;
            v2f bf; bf.x = sW[krow][ncol]; bf.y = sW[krow + 1][ncol];
            bfrag[kk] = bf;
        }

        const size_t rowbase = (size_t)blockIdx.x * MROWS;
        #pragma unroll
        for (int m = 0; m < 8; ++m) {
            int rl = m * 16 + lm;
            if (rl > MROWS - 1) rl = MROWS - 1;   // clamp (results discarded below)
            const v2f* arow = (const v2f*)(x_sources + (rowbase + (size_t)rl) * NBAND);
            v8f acc = {0.f, 0.f, 0.f, 0.f, 0.f, 0.f, 0.f, 0.f};
            #pragma unroll
            for (int kk = 0; kk < 4; ++kk) {
                v2f afrag = arow[2 * kk + half];  // K = 4*kk + 2*half .. +1
                acc = __builtin_amdgcn_wmma_f32_16x16x4_f32(
                          false, afrag, false, bfrag[kk],
                          (short)0, acc, false, false);
            }
            #pragma unroll
            for (int v = 0; v < 8; ++v) {
                const int row = m * 16 + v + 8 * half;   // C/D layout: M = v + 8*half
                if (row < MROWS) sXS[row][ncol] = acc[v];
            }
        }
    }
    __syncthreads();

    // ---- Phase 2: per-(b,c) two-stream scans ----
    const int    b_local = tid / NCH;
    const int    c       = tid - b_local * NCH;
    const size_t b       = (size_t)blockIdx.x * BPB + b_local;

    const float*  xsr = &sXS[b_local * NROWS][0];     // hl[i] = xsr[i*NCH + c]
    const size_t  ibase = (b * NLAYER) * NCH + c;
    const float3* es3 = (const float3*)e_split;

    const float emis = x_emis[b];
    float rs_cur = 1.f - emis;                        // rs[60] = r_surface
    float ss_cur = xsr[61 * NCH + c] * emis;          // ss[60] = surface_source * emissivity
    float hl_hi  = xsr[60 * NCH + c];

    float ts[NLAYER], Aa[NLAYER], Rs[NLAYER], Ss[NLAYER];

    // Upward (reverse) scan, l = 59 .. 0, streaming inputs once
    #pragma unroll
    for (int l = NLAYER - 1; l >= 0; --l) {
        const size_t li   = ibase + (size_t)l * NCH;
        const float  tfv  = t_full[li];
        const float3 ev   = es3[li];
        const float  tauv = tau[li];
        const float  hl_lo = xsr[l * NCH + c];

        const float e = 1.f - tfv;
        const float t = tfv + e * ev.x;
        const float r = e * ev.y;
        const float a = e * ev.z;

        const float coef = (hl_hi - hl_lo) / (tauv + 1e-6f);
        float s_up = (coef + hl_lo) - r * (hl_lo - coef) - t * (coef + hl_hi);
        float s_dn = (hl_hi - coef) - r * (coef + hl_hi) - t * (hl_lo - coef);
        const float s_both = a * 0.5f * (hl_hi + hl_lo);
        const bool thick = tauv > 0.001f;
        s_up = thick ? s_up : s_both;
        s_dn = thick ? s_dn : s_both;

        const float dd = 1.f / (1.f - rs_cur * r);

        ts[l] = t;
        Rs[l] = rs_cur;                       // rs[l+1]
        Ss[l] = ss_cur;                       // ss[l+1]
        Aa[l] = r * ss_cur + s_dn * dd;       // r[l]*ss[l+1] + os_dn[l]*ds[l]

        const float ss_new = s_up + t * (ss_cur + rs_cur * s_dn) * dd;
        const float rs_new = r + t * t * rs_cur * dd;
        ss_cur = ss_new;
        rs_cur = rs_new;
        hl_hi  = hl_lo;
    }

    // Downward scan, l = 0 .. 59, direct coalesced stores
    const size_t obase = (b * (NLAYER + 1)) * NCH + c;
    flux_down[obase] = 0.f;
    flux_up[obase]   = ss_cur;                // ss[0]
    float fd = 0.f;
    #pragma unroll
    for (int l = 0; l < NLAYER; ++l) {
        fd = ts[l] * fd + Aa[l];
        const float fu = fd * Rs[l] + Ss[l];
        flux_down[obase + (size_t)(l + 1) * NCH] = fd;
        flux_up[obase + (size_t)(l + 1) * NCH]   = fu;
    }
}

extern "C" void kernel_launch(void* const* d_in, const int* in_sizes, int n_in,
                              void* d_out, int out_size, void* d_ws, size_t ws_size,
                              hipStream_t stream) {
    const float* x_sources = (const float*)d_in[0];
    const float* t_full    = (const float*)d_in[1];
    const float* e_split   = (const float*)d_in[2];
    const float* tau       = (const float*)d_in[3];
    const float* x_emis    = (const float*)d_in[4];
    const float* b2c       = (const float*)d_in[5];

    float* fd = (float*)d_out;
    float* fu = fd + (size_t)NBATCH * (NLAYER + 1) * NCH;

    dim3 grid(NBATCH / BPB);
    dim3 block(NTHREADS);
    hipLaunchKernelGGL(radiative_scan_kernel, grid, block, 0, stream,
                       x_sources, t_full, e_split, tau, x_emis, b2c, fd, fu);
}